// LennardJonesK_36206574305301
// MI455X (gfx1250) — compile-verified
//
#include <hip/hip_runtime.h>

typedef float v2f __attribute__((ext_vector_type(2)));
typedef float v8f __attribute__((ext_vector_type(8)));

#define N_ROOT 512
#define N_IMG  27
#define M_TOT  (N_ROOT * N_IMG)   // 13824
#define N_SPLIT 27                // 864 j-tiles / 32 per wave
#define JT_PER_WAVE 32
#define J_SLICE 512               // 32 tiles * 16

// ---------------- ws layout (bytes) ----------------
#define XSQ_OFF    0                                      // float4[13824]  = 221184
#define ROOTQ_OFF  (XSQ_OFF + M_TOT * 16)                 // float4[512]    = 8192
#define FPART_OFF  (ROOTQ_OFF + N_ROOT * 16)              // float[27*512*3]= 165888
#define EPART_OFF  (FPART_OFF + N_SPLIT * N_ROOT * 3 * 4) // float[864]     = 3456

#if defined(__HIP_DEVICE_COMPILE__) && \
    __has_builtin(__builtin_amdgcn_global_load_async_to_lds_b128) && \
    __has_builtin(__builtin_amdgcn_s_wait_asynccnt)
#define HAVE_ASYNC_LDS 1
#define AS1 __attribute__((address_space(1)))
#define AS3 __attribute__((address_space(3)))
typedef int vi4 __attribute__((vector_size(16)));   // b128 payload type per builtin signature
#endif

// Build cartesian image positions xs[k] = (frac[id] + offset[k]) @ L, and root positions.
__global__ __launch_bounds__(256) void lj_build_xs(
    const float* __restrict__ pos, const float* __restrict__ lat,
    const float* __restrict__ off, const int* __restrict__ ids,
    float4* __restrict__ xsq, float4* __restrict__ rootq) {
  int k = blockIdx.x * 256 + threadIdx.x;
  if (k >= M_TOT) return;
  int a = ids[k];
  float f0 = pos[3 * a + 0] + off[3 * k + 0];
  float f1 = pos[3 * a + 1] + off[3 * k + 1];
  float f2 = pos[3 * a + 2] + off[3 * k + 2];
  // row-vector @ row-major L:  out[i] = sum_r f[r] * L[r*3 + i]
  float x = f0 * lat[0] + f1 * lat[3] + f2 * lat[6];
  float y = f0 * lat[1] + f1 * lat[4] + f2 * lat[7];
  float z = f0 * lat[2] + f1 * lat[5] + f2 * lat[8];
  xsq[k] = make_float4(x, y, z, 0.0f);
  if (k % N_IMG == 13) rootq[a] = make_float4(x, y, z, 0.0f);  // offset (0,0,0) image
}

// Main pair kernel.
// Grid: 108 blocks x 256 threads. Block -> (split = bid/4, 8 atom-tiles = (bid%4)*8 + wave).
// Each wave: 16 root atoms x 512 images (its split slice), staged once in LDS per block.
// Lane mapping (A-matrix layout of v_wmma_f32_16x16x4_f32):
//   row = lane&15 (atom M), hi = lane>>4; slot s -> j-column 4*(s>>1) + (s&1) + 2*hi.
__global__ __launch_bounds__(256) void lj_pairs(
    const float4* __restrict__ xsq, const float4* __restrict__ rootq,
    float* __restrict__ Fpart, float* __restrict__ Epart) {
  __shared__ float4 sQ[J_SLICE];
  __shared__ float sT[8][16][3];

  const int tid = threadIdx.x;
  const int w = tid >> 5;
  const int l = tid & 31;
  const int split = blockIdx.x >> 2;                 // 0..26
  const int atile = ((blockIdx.x & 3) << 3) + w;     // 0..31
  const int row = l & 15;
  const int hi = l >> 4;
  const int jSlice = split * J_SLICE;

  // Stage this split's 512 images (8 KB) into LDS, shared by all 8 waves.
#ifdef HAVE_ASYNC_LDS
  for (int k = tid; k < J_SLICE; k += 256) {
    __builtin_amdgcn_global_load_async_to_lds_b128(
        (AS1 vi4*)&xsq[jSlice + k], (AS3 vi4*)&sQ[k], 0, 0);
  }
  __builtin_amdgcn_s_wait_asynccnt(0);   // own async loads landed in LDS
#else
  for (int k = tid; k < J_SLICE; k += 256) sQ[k] = xsq[jSlice + k];
#endif
  __syncthreads();                        // all waves' loads visible

  const int a = atile * 16 + row;
  const float4 p = rootq[a];
  const int selfJ = a * N_IMG + 13;

  float e_acc = 0.0f;
  float csum = 0.0f;
  v8f T = {0.0f, 0.0f, 0.0f, 0.0f, 0.0f, 0.0f, 0.0f, 0.0f};

  for (int jt = 0; jt < JT_PER_WAVE; ++jt) {
    const int lB = jt << 4;
    const int jB = jSlice + lB;

    // Phase 1: batch all 8 LDS loads (broadcast within each lane half).
    float4 q[8];
#pragma unroll
    for (int s = 0; s < 8; ++s) {
      const int jcol = ((s >> 1) << 2) + (s & 1) + (hi << 1);
      q[s] = sQ[lB + jcol];
    }

    // Phase 2: pair math.
    float coef[8];
    float bsel[8];
#pragma unroll
    for (int s = 0; s < 8; ++s) {
      const int jcol = ((s >> 1) << 2) + (s & 1) + (hi << 1);
      const float dx = p.x - q[s].x, dy = p.y - q[s].y, dz = p.z - q[s].z;
      float d2 = __builtin_fmaf(dx, dx, __builtin_fmaf(dy, dy, dz * dz));
      const bool self = (jB + jcol) == selfJ;
      d2 = self ? 1.0f : d2;               // avoids inf/NaN; e-term becomes exact 0
      const float inv = __builtin_amdgcn_rcpf(d2);
      const float inv2 = inv * inv;
      const float c6 = inv2 * inv;
      const float c12 = c6 * c6;
      const float e = c12 - c6;            // exactly 0 for self (d2==1)
      e_acc += e;
      float kk = (c12 + e) * inv;          // (2*c12 - c6) / d2
      kk = self ? 0.0f : kk;
      csum += kk;
      coef[s] = kk;
      // B operand: component 'row' of q (columns N>=3 are zero)
      bsel[s] = (row == 0) ? q[s].x : (row == 1) ? q[s].y : (row == 2) ? q[s].z : 0.0f;
    }

    // T[16x16] += coef[16x16] @ qcomp[16x16] as 4 chained 16x16x4 f32 WMMAs.
#pragma unroll
    for (int c = 0; c < 4; ++c) {
      v2f av; av.x = coef[2 * c]; av.y = coef[2 * c + 1];
      v2f bv; bv.x = bsel[2 * c]; bv.y = bsel[2 * c + 1];
      T = __builtin_amdgcn_wmma_f32_16x16x4_f32(false, av, false, bv,
                                                (short)0, T, false, false);
    }
  }

  // ---- wave-level wrap-up ----
#pragma unroll
  for (int o = 16; o >= 1; o >>= 1) e_acc += __shfl_xor(e_acc, o);
  csum += __shfl_xor(csum, 16);            // full row sum over 16 columns

  if (row < 3) {                           // D element (M = v + 8*hi, N = row)
#pragma unroll
    for (int v = 0; v < 8; ++v) sT[w][v + 8 * hi][row] = T[v];
  }
  __syncthreads();

  if (l < 16) {
    const float tx = sT[w][row][0];
    const float ty = sT[w][row][1];
    const float tz = sT[w][row][2];
    float* fp = Fpart + (split * N_ROOT + a) * 3;
    fp[0] = 24.0f * (csum * p.x - tx);
    fp[1] = 24.0f * (csum * p.y - ty);
    fp[2] = 24.0f * (csum * p.z - tz);
  }
  if (l == 0) Epart[split * 32 + atile] = e_acc;
}

// Deterministic final reduction: forces over 27 splits, energy over 864 wave partials.
__global__ __launch_bounds__(1024) void lj_finalize(
    const float* __restrict__ Fpart, const float* __restrict__ Epart,
    float* __restrict__ out) {
  __shared__ float red[1024];
  const int tid = threadIdx.x;
  if (tid < N_ROOT) {
    float fx = 0.0f, fy = 0.0f, fz = 0.0f;
    for (int s = 0; s < N_SPLIT; ++s) {
      const float* fp = Fpart + (s * N_ROOT + tid) * 3;
      fx += fp[0]; fy += fp[1]; fz += fp[2];
    }
    out[1 + 3 * tid + 0] = fx;
    out[1 + 3 * tid + 1] = fy;
    out[1 + 3 * tid + 2] = fz;
  }
  const int NW = N_SPLIT * 32;  // 864
  red[tid] = (tid < NW) ? Epart[tid] : 0.0f;
  __syncthreads();
  for (int st = 512; st > 0; st >>= 1) {
    if (tid < st) red[tid] += red[tid + st];
    __syncthreads();
  }
  if (tid == 0) out[0] = 2.0f * red[0];    // 0.5 * 4 * sum(c12 - c6)
}

extern "C" void kernel_launch(void* const* d_in, const int* in_sizes, int n_in,
                              void* d_out, int out_size, void* d_ws, size_t ws_size,
                              hipStream_t stream) {
  (void)in_sizes; (void)n_in; (void)out_size; (void)ws_size;
  const float* pos = (const float*)d_in[0];   // [512,3] fractional
  const float* lat = (const float*)d_in[1];   // [3,3]
  const float* off = (const float*)d_in[2];   // [13824,3]
  const int* ids   = (const int*)d_in[3];     // [13824]
  char* ws = (char*)d_ws;
  float4* xsq   = (float4*)(ws + XSQ_OFF);
  float4* rootq = (float4*)(ws + ROOTQ_OFF);
  float* Fpart  = (float*)(ws + FPART_OFF);
  float* Epart  = (float*)(ws + EPART_OFF);
  float* out    = (float*)d_out;

  lj_build_xs<<<(M_TOT + 255) / 256, 256, 0, stream>>>(pos, lat, off, ids, xsq, rootq);
  lj_pairs<<<N_SPLIT * 4, 256, 0, stream>>>(xsq, rootq, Fpart, Epart);
  lj_finalize<<<1, 1024, 0, stream>>>(Fpart, Epart, out);
}